// PositionalEncoding_386547056977
// MI455X (gfx1250) — compile-verified
//
#include <hip/hip_runtime.h>

// Reference: x [8, 4096, 1024] fp32; out = x + pe(seq, dim) broadcast over batch.
// pe[s,d] = sin/cos( s / 10000^( (d//2) / 1024 ) )  (exponent k/dim, faithful to source)

#define SEQ_LEN 4096
#define EMB_DIM 1024
#define NBATCH  8

typedef __attribute__((ext_vector_type(4))) float v4f;

__global__ __launch_bounds__(256) void pe_add_async_kernel(
    const float* __restrict__ x, float* __restrict__ out) {
  const int s = blockIdx.x;   // sequence row 0..4095
  const int t = threadIdx.x;  // 0..255 ; covers D=1024 as float4 per thread

  __shared__ float tile[NBATCH * EMB_DIM];  // 32 KiB: this row for all 8 batches

  const size_t rowOff      = (size_t)s * EMB_DIM + 4u * (unsigned)t; // elem offset in one batch
  const size_t batchStride = (size_t)SEQ_LEN * EMB_DIM;

  // ---- 1) Kick off async DMA global -> LDS for all 8 batch copies of this row.
  //      Tracked by ASYNCcnt; per-lane 16B transfer (b128).
#pragma unroll
  for (int b = 0; b < NBATCH; ++b) {
    const float* g = x + (size_t)b * batchStride + rowOff;
    // Low 32 bits of a generic shared pointer == wave-relative LDS byte address
    // (flat->LDS aperture truncation rule), which is what VDST expects.
    unsigned lds = (unsigned)(size_t)&tile[b * EMB_DIM + 4 * t];
    asm volatile("global_load_async_to_lds_b128 %0, %1, off"
                 :: "v"(lds), "v"(g)
                 : "memory");
  }

  // ---- 2) Compute the positional encoding for this thread's 4 columns while
  //         the async copies are in flight (latency hiding).
  //      d0=4t   -> k=2t   (even -> sin)
  //      d1=4t+1 -> k=2t   (odd  -> cos)
  //      d2=4t+2 -> k=2t+1 (even -> sin)
  //      d3=4t+3 -> k=2t+1 (odd  -> cos)
  const float c  = -1.29762816206537187e-2f;  // -log2(10000) / 1024
  const float k0 = (float)(2 * t);
  const float k1 = k0 + 1.0f;
  const float pos = (float)s;
  const float a0 = pos * __builtin_exp2f(k0 * c);  // s * 10000^(-k0/1024)
  const float a1 = pos * __builtin_exp2f(k1 * c);
  v4f pe;
  pe.x = __sinf(a0);
  pe.y = __cosf(a0);
  pe.z = __sinf(a1);
  pe.w = __cosf(a1);

  // ---- 3) Wait for the DMA, then add + stream out with non-temporal stores.
  //      Each thread reads exactly the LDS bytes it requested -> no barrier needed.
  asm volatile("s_wait_asynccnt 0" ::: "memory");

#pragma unroll
  for (int b = 0; b < NBATCH; ++b) {
    v4f v = *(const v4f*)&tile[b * EMB_DIM + 4 * t];
    v.x += pe.x;
    v.y += pe.y;
    v.z += pe.z;
    v.w += pe.w;
    v4f* o = (v4f*)(out + (size_t)b * batchStride + rowOff);
    __builtin_nontemporal_store(v, o);  // write-once output: bypass near caches
  }
}

extern "C" void kernel_launch(void* const* d_in, const int* in_sizes, int n_in,
                              void* d_out, int out_size, void* d_ws, size_t ws_size,
                              hipStream_t stream) {
  (void)in_sizes; (void)n_in; (void)d_ws; (void)ws_size; (void)out_size;
  const float* x  = (const float*)d_in[0];
  float*       out = (float*)d_out;

  dim3 grid(SEQ_LEN);   // one block per sequence row; PE computed once, reused x8 batches
  dim3 block(256);      // 8 wave32 waves; 256 * float4 = 1024 = EMB_DIM
  hipLaunchKernelGGL(pe_add_async_kernel, grid, block, 0, stream, x, out);
}